// Partition_enhanced_GCN_21449066676827
// MI455X (gfx1250) — compile-verified
//
#include <hip/hip_runtime.h>
#include <hip/hip_bf16.h>

// ---------------- problem constants (match reference) ----------------
#define NN   50000   // nodes
#define NE   800000  // edges
#define IND  32      // input feature dim
#define HD   64      // hidden channels
#define OUTD 32      // output classes
#define LL   3       // num layers
#define CC   8       // num clusters
#define GG   64      // graphs in batch

typedef float v2f __attribute__((ext_vector_type(2)));
typedef float v8f __attribute__((ext_vector_type(8)));

// ---------------- degree / normalization ----------------
__global__ void k_init_deg(float* deg, int n) {
    int i = blockIdx.x * blockDim.x + threadIdx.x;
    if (i < n) deg[i] = 1.0f;                        // self loop contributes 1
}

__global__ void k_count_deg(const int* __restrict__ col, float* deg, int e) {
    int i = blockIdx.x * blockDim.x + threadIdx.x;
    if (i < e) atomicAdd(&deg[col[i]], 1.0f);
}

__global__ void k_dinv(float* deg, int n) {
    int i = blockIdx.x * blockDim.x + threadIdx.x;
    if (i < n) {
        float d = deg[i];
        deg[i] = (d > 0.0f) ? rsqrtf(d) : 0.0f;      // in-place deg -> dinv
    }
}

// ---------------- aggregation: a = A_norm @ h ----------------
// self-loop term also zero-initializes the accumulation buffer:
// a[i,f] = dinv[i]^2 * h[i,f]
__global__ void k_selfloop(const float* __restrict__ h, const float* __restrict__ dinv,
                           float* __restrict__ a, int n, int d) {
    int i = blockIdx.x * blockDim.x + threadIdx.x;
    if (i < n * d) {
        int node = i / d;
        float w = dinv[node];
        a[i] = w * w * h[i];
    }
}

// one wave per edge; lanes stride the feature dim
__global__ void k_edge_scatter(const int* __restrict__ row, const int* __restrict__ col,
                               const float* __restrict__ dinv, const float* __restrict__ h,
                               float* __restrict__ a, int e, int d) {
    int gid  = blockIdx.x * blockDim.x + threadIdx.x;
    int lane = gid & 31;
    int eidx = gid >> 5;
    if (eidx >= e) return;
    int r = row[eidx];
    int c = col[eidx];
    float w = dinv[r] * dinv[c];
    const float* hr = h + (size_t)r * d;
    float*       ac = a + (size_t)c * d;
    for (int f = lane; f < d; f += 32)
        atomicAdd(&ac[f], w * hr[f]);
}

// ---------------- per-cluster GEMM + ReLU via V_WMMA_F32_16X16X4_F32 -------
// Hout[n, :HD] = relu( a[n, :K] @ W[labels[n]] ),  W: [CC, K, HD] row-major.
// One wave computes one 16(node) x 16(out) tile; loops only over clusters
// present in the tile; per-row select by label afterwards.
template <int K>
__global__ void k_cluster_gemm(const float* __restrict__ A, const float* __restrict__ W,
                               const int* __restrict__ labels, float* __restrict__ Hout,
                               int n) {
    const int KSTEPS = K / 4;
    int wave = (blockIdx.x * blockDim.x + threadIdx.x) >> 5;
    int lane = threadIdx.x & 31;
    int ntm  = (n + 15) >> 4;
    int tile_m = wave >> 2;            // HD/16 == 4 output tiles per node tile
    int tile_n = wave & 3;
    if (tile_m >= ntm) return;         // wave-uniform exit (EXEC stays all-1s)

    int m0   = tile_m << 4;
    int o0   = tile_n << 4;
    int half = lane >> 4;              // 0: K pair {0,1}; 1: K pair {2,3}
    int l16  = lane & 15;

    // --- A fragments for every K-step (kept in registers: K is template) ---
    int  arow   = m0 + l16;
    bool avalid = arow < n;
    int  arowc  = avalid ? arow : 0;
    const float* Arow = A + (size_t)arowc * K;
    v2f afrag[KSTEPS];
#pragma unroll
    for (int ks = 0; ks < KSTEPS; ++ks) {
        int k = (ks << 2) + (half << 1);
        afrag[ks].x = avalid ? Arow[k]     : 0.0f;
        afrag[ks].y = avalid ? Arow[k + 1] : 0.0f;
    }

    // --- labels for the 8 D-rows this lane owns; cluster presence mask ---
    int labs[8];
    unsigned use = 0;
#pragma unroll
    for (int r = 0; r < 8; ++r) {
        int node = m0 + r + (half << 3);
        int lb = (node < n) ? labels[node] : -1;
        labs[r] = lb;
        if (lb >= 0) use |= (1u << lb);
    }
#pragma unroll
    for (int off = 16; off > 0; off >>= 1)
        use |= (unsigned)__shfl_xor((int)use, off, 32);
    use = (unsigned)__builtin_amdgcn_readfirstlane((int)use);  // scalar branch

    v8f res = {};
    for (int c = 0; c < CC; ++c) {
        if (!((use >> c) & 1u)) continue;
        const float* Wc = W + (size_t)c * K * HD;
        v8f acc = {};
#pragma unroll
        for (int ks = 0; ks < KSTEPS; ++ks) {
            int k = (ks << 2) + (half << 1);
            v2f b;
            b.x = Wc[(size_t)k       * HD + o0 + l16];
            b.y = Wc[(size_t)(k + 1) * HD + o0 + l16];
            acc = __builtin_amdgcn_wmma_f32_16x16x4_f32(
                false, afrag[ks], false, b, (short)0, acc, false, false);
        }
#pragma unroll
        for (int r = 0; r < 8; ++r)
            if (labs[r] == c) res[r] = acc[r];
    }

    // --- ReLU + store ---
#pragma unroll
    for (int r = 0; r < 8; ++r) {
        int node = m0 + r + (half << 3);
        if (node < n) {
            float v = res[r];
            Hout[(size_t)node * HD + o0 + l16] = v > 0.0f ? v : 0.0f;
        }
    }
}

// ---------------- pooling + readout MLP ----------------
__global__ void k_zero(float* p, int count) {
    int i = blockIdx.x * blockDim.x + threadIdx.x;
    if (i < count) p[i] = 0.0f;
}

__global__ void k_pool(const float* __restrict__ h, const int* __restrict__ batch,
                       float* __restrict__ xp, int n, int t) {
    int i = blockIdx.x * blockDim.x + threadIdx.x;
    if (i < n * HD) {
        int node = i / HD;
        int o    = i - node * HD;
        int g    = batch[node];
        atomicAdd(&xp[(size_t)g * (HD * LL) + t * HD + o], h[i]);
    }
}

__global__ void k_mlp1(const float* __restrict__ xp, const float* __restrict__ W1,
                       const float* __restrict__ b1, float* __restrict__ hid) {
    int i = blockIdx.x * blockDim.x + threadIdx.x;
    if (i < GG * HD) {
        int g = i / HD, o = i - (i / HD) * HD;
        float s = b1[o];
        const float* xr = xp + (size_t)g * (HD * LL);
        for (int k = 0; k < HD * LL; ++k) s += xr[k] * W1[k * HD + o];
        hid[i] = s > 0.0f ? s : 0.0f;
    }
}

__global__ void k_mlp2(const float* __restrict__ hid, const float* __restrict__ W2,
                       const float* __restrict__ b2, float* __restrict__ out) {
    int i = blockIdx.x * blockDim.x + threadIdx.x;
    if (i < GG * OUTD) {
        int g = i / OUTD, o = i - (i / OUTD) * OUTD;
        float s = b2[o];
        const float* hr = hid + (size_t)g * HD;
        for (int k = 0; k < HD; ++k) s += hr[k] * W2[k * OUTD + o];
        out[i] = s;
    }
}

// ---------------- orchestration ----------------
extern "C" void kernel_launch(void* const* d_in, const int* in_sizes, int n_in,
                              void* d_out, int out_size, void* d_ws, size_t ws_size,
                              hipStream_t stream) {
    const float* x       = (const float*)d_in[0];
    const int*   labels  = (const int*)  d_in[1];
    const int*   eidx    = (const int*)  d_in[2];   // [2, E]
    const int*   batch   = (const int*)  d_in[3];
    const float* W_first = (const float*)d_in[4];   // [C, IN, H]
    const float* W_rest  = (const float*)d_in[5];   // [L-1, C, H, H]
    const float* W1      = (const float*)d_in[6];   // [H*L, H]
    const float* b1      = (const float*)d_in[7];
    const float* W2      = (const float*)d_in[8];   // [H, OUT]
    const float* b2      = (const float*)d_in[9];
    float*       out     = (float*)d_out;

    const int* row = eidx;
    const int* col = eidx + NE;

    // workspace layout (floats)
    float* ws   = (float*)d_ws;
    float* dinv = ws;                                   // N
    float* Abuf = ws + ((NN + 255) & ~255);             // N*HD
    float* Hbuf = Abuf + (size_t)NN * HD;               // N*HD
    float* xp   = Hbuf + (size_t)NN * HD;               // G*HD*L
    float* hid  = xp + GG * HD * LL;                    // G*HD

    const int T = 256;

    // normalization
    k_init_deg <<<(NN + T - 1) / T, T, 0, stream>>>(dinv, NN);
    k_count_deg<<<(NE + T - 1) / T, T, 0, stream>>>(col, dinv, NE);
    k_dinv     <<<(NN + T - 1) / T, T, 0, stream>>>(dinv, NN);
    k_zero     <<<(GG * HD * LL + T - 1) / T, T, 0, stream>>>(xp, GG * HD * LL);

    int nwaves  = ((NN + 15) / 16) * 4;          // 16x16 tiles, HD/16 = 4
    int gblocks = (nwaves + 7) / 8;              // 8 waves per 256-thread block

    for (int t = 0; t < LL; ++t) {
        int d = (t == 0) ? IND : HD;
        const float* hin = (t == 0) ? x : Hbuf;
        const float* Wt  = (t == 0) ? W_first : (W_rest + (size_t)(t - 1) * CC * HD * HD);

        long nsl = (long)NN * d;
        k_selfloop<<<(int)((nsl + T - 1) / T), T, 0, stream>>>(hin, dinv, Abuf, NN, d);

        long nes = (long)NE * 32;                // one wave per edge
        k_edge_scatter<<<(int)((nes + T - 1) / T), T, 0, stream>>>(row, col, dinv, hin,
                                                                   Abuf, NE, d);
        if (t == 0)
            k_cluster_gemm<IND><<<gblocks, T, 0, stream>>>(Abuf, Wt, labels, Hbuf, NN);
        else
            k_cluster_gemm<HD> <<<gblocks, T, 0, stream>>>(Abuf, Wt, labels, Hbuf, NN);

        long npl = (long)NN * HD;
        k_pool<<<(int)((npl + T - 1) / T), T, 0, stream>>>(Hbuf, batch, xp, NN, t);
    }

    k_mlp1<<<(GG * HD + T - 1) / T, T, 0, stream>>>(xp, W1, b1, hid);
    k_mlp2<<<(GG * OUTD + T - 1) / T, T, 0, stream>>>(hid, W2, b2, out);
}